// DeformConv_12360915878479
// MI455X (gfx1250) — compile-verified
//
#include <hip/hip_runtime.h>
#include <math.h>

// ---------------------------------------------------------------------------
// DeformConv (RegSeg) for MI455X / gfx1250.
//  K0: NCHW -> N(HW)C transpose of x and x2 (coalesced bilinear gathers later)
//  K0b: repack deform_w -> w_t[k][c/2][oc][2] (K-pairs adjacent -> b64 A-frag
//       loads), offset/mask w -> w1_t[k][c][27]
//  K1: fused offset+mask 3x3 conv (sigmoid on mask), LDS-staged weights
//  K2: fused modulated bilinear sampling (into LDS, K-paired layout) +
//      fp32 WMMA GEMM via v_wmma_f32_16x16x4_f32 (exact fp32 semantics).
//      B fragments are single ds_load_b64, A fragments single global_load_b64.
// ---------------------------------------------------------------------------

typedef __attribute__((ext_vector_type(2))) float v2f;
typedef __attribute__((ext_vector_type(8))) float v8f;

// Guard only on the DEVICE pass: host clang doesn't declare amdgcn builtins.
#if defined(__HIP_DEVICE_COMPILE__) && \
    !__has_builtin(__builtin_amdgcn_wmma_f32_16x16x4_f32)
#error "missing __builtin_amdgcn_wmma_f32_16x16x4_f32 on this toolchain"
#endif

#define B_   4
#define C_   128
#define H_   96
#define W_   96
#define HW_  (H_ * W_)
#define O_   128
#define K_   9
#define NOFF 18
#define NCH  27    // 18 offset + 9 mask channels
// LDS val tile: val[cp][px][2], cp = c/2 (64 rows), 64 px, pair-row pitch 160
// floats (128 data + 32 pad) => rows cp and cp+1 hit disjoint bank halves
// (160 % 64 == 32), so full-wave b64 accesses are bank-conflict-free.
#define PP   160

// ---------------- NCHW -> N(HW)C tiled transpose ---------------------------
__global__ void transpose_chw_hwc(const float* __restrict__ in,
                                  float* __restrict__ out) {
  __shared__ float t[32][33];
  const int b    = blockIdx.z;
  const int pixT = blockIdx.x * 32;
  const int cT   = blockIdx.y * 32;
#pragma unroll
  for (int i = 0; i < 4; ++i) {
    int c = cT + threadIdx.y + i * 8;
    t[threadIdx.y + i * 8][threadIdx.x] =
        in[((size_t)b * C_ + c) * HW_ + pixT + threadIdx.x];
  }
  __syncthreads();
#pragma unroll
  for (int i = 0; i < 4; ++i) {
    int pix = pixT + threadIdx.y + i * 8;
    out[((size_t)b * HW_ + pix) * C_ + cT + threadIdx.x] =
        t[threadIdx.x][threadIdx.y + i * 8];
  }
}

// ---------------- weight repack -------------------------------------------
// w_t[k][cp][oc][i] = deform_w[oc][2*cp+i][k]  (K-pair contiguous per oc)
// w1_t[k][c][j]     = offset_w[j][c][k] (j<18) else mask_w[j-18][c][k]
__global__ void reorg_weights(const float* __restrict__ dw,
                              const float* __restrict__ ow,
                              const float* __restrict__ mw,
                              float* __restrict__ w_t,
                              float* __restrict__ w1_t) {
  int idx = blockIdx.x * blockDim.x + threadIdx.x;
  if (idx < K_ * C_ * O_) {
    int k  = idx / (C_ * O_);
    int cp = (idx / 256) % (C_ / 2);
    int oc = (idx >> 1) & (O_ - 1);
    int i  = idx & 1;
    int c  = 2 * cp + i;
    w_t[idx] = dw[((size_t)oc * C_ + c) * K_ + k];
  }
  if (idx < K_ * C_ * NCH) {
    int k = idx / (C_ * NCH);
    int c = (idx / NCH) % C_;
    int j = idx % NCH;
    w1_t[idx] = (j < NOFF) ? ow[((size_t)j * C_ + c) * K_ + k]
                           : mw[((size_t)(j - NOFF) * C_ + c) * K_ + k];
  }
}

// ---------------- fused offset + mask conv ---------------------------------
// offmask[b][j][pix]: j<18 -> offset (k*2 + {dy,dx}), j>=18 -> sigmoid(mask_k)
__global__ void offset_mask_conv(const float* __restrict__ x2t,   // [B][HW][C]
                                 const float* __restrict__ w1_t,  // [K][C][27]
                                 const float* __restrict__ ob,
                                 const float* __restrict__ mb,
                                 float* __restrict__ offmask) {   // [B][27][HW]
  __shared__ float wsh[C_ * NCH];  // 13824 B
  const int b   = blockIdx.x / (HW_ / 256);
  const int pix = (blockIdx.x % (HW_ / 256)) * 256 + threadIdx.x;
  const int oy  = pix / W_, ox = pix % W_;

  float acc[NCH];
#pragma unroll
  for (int j = 0; j < NCH; ++j) acc[j] = 0.f;

  const float* xb = x2t + (size_t)b * HW_ * C_;
  for (int k = 0; k < K_; ++k) {
    __syncthreads();
    for (int i = threadIdx.x; i < C_ * NCH; i += 256)
      wsh[i] = w1_t[k * C_ * NCH + i];
    __syncthreads();
    const int yy = oy - 1 + k / 3;
    const int xx = ox - 1 + k % 3;
    if (yy >= 0 && yy < H_ && xx >= 0 && xx < W_) {
      const float* row = xb + ((size_t)yy * W_ + xx) * C_;
      for (int c = 0; c < C_; ++c) {
        const float xv = row[c];
        const float* wr = &wsh[c * NCH];
#pragma unroll
        for (int j = 0; j < NCH; ++j) acc[j] += xv * wr[j];
      }
    }
  }
  float* om = offmask + (size_t)b * NCH * HW_;
#pragma unroll
  for (int j = 0; j < NOFF; ++j) om[j * HW_ + pix] = acc[j] + ob[j];
#pragma unroll
  for (int j = 0; j < K_; ++j) {
    float v = acc[NOFF + j] + mb[j];
    om[(NOFF + j) * HW_ + pix] = 1.f / (1.f + expf(-v));
  }
}

// ---------------- fused sampling + fp32 WMMA GEMM --------------------------
// One workgroup (256 thr = 8 waves) -> 128 oc x 64 pixels for one batch.
// Wave w: oc in [16w,16w+16), all 64 pixels (4 WMMA N-subtiles).
// Sampling split: csub = wave&3 picks channel quarter, pid = lane + 32*(wave/4)
// picks pixel -> per-wave lanes write 64 consecutive floats per pair-row
// (conflict-free ds_store_b64) and offset reads are lane-coalesced.
__global__ void __launch_bounds__(256) deform_wmma(
    const float* __restrict__ xt,       // [B][HW][C]
    const float* __restrict__ offmask,  // [B][27][HW]
    const float* __restrict__ w_t,      // [K][C/2][O][2]
    const float* __restrict__ db,       // [O]
    float* __restrict__ out) {          // [B][O][HW]
  __shared__ __align__(16) float val[(C_ / 2) * PP];  // 40960 B

  const int tid     = threadIdx.x;
  const int b       = blockIdx.x / (HW_ / 64);
  const int pixBase = (blockIdx.x % (HW_ / 64)) * 64;
  const int lane = tid & 31, wave = tid >> 5;
  const int hi = lane >> 4, lo = lane & 15;
  const int ocBase = wave * 16;

  // sampling work split
  const int csub = wave & 3;
  const int c0   = csub * 32;
  const int pid  = lane + ((wave >> 2) << 5);  // 0..63
  const int pix  = pixBase + pid;
  const int oy = pix / W_, ox = pix % W_;
  const float* xb = xt + (size_t)b * HW_ * C_;
  const float* om = offmask + (size_t)b * NCH * HW_;

  v8f acc[4] = {};  // 16 oc x 64 px accumulators, persist across taps

  for (int k = 0; k < K_; ++k) {
    // ---- gather / modulated-bilinear phase ----
    const float dy = om[(2 * k) * HW_ + pix];
    const float dx = om[(2 * k + 1) * HW_ + pix];
    const float m  = om[(NOFF + k) * HW_ + pix];
    const float py = (float)(oy - 1 + k / 3) + dy;
    const float px = (float)(ox - 1 + k % 3) + dx;
    const float y0f = floorf(py), x0f = floorf(px);
    const int y0 = (int)y0f, x0 = (int)x0f;
    const int y1 = y0 + 1, x1 = x0 + 1;
    const float ly = py - y0f, lx = px - x0f;
    const float vy0 = (y0 >= 0 && y0 < H_) ? 1.f : 0.f;
    const float vy1 = (y1 >= 0 && y1 < H_) ? 1.f : 0.f;
    const float vx0 = (x0 >= 0 && x0 < W_) ? 1.f : 0.f;
    const float vx1 = (x1 >= 0 && x1 < W_) ? 1.f : 0.f;
    const float w00 = (1.f - ly) * (1.f - lx) * m * vy0 * vx0;
    const float w01 = (1.f - ly) * lx * m * vy0 * vx1;
    const float w10 = ly * (1.f - lx) * m * vy1 * vx0;
    const float w11 = ly * lx * m * vy1 * vx1;
    const int cy0 = min(max(y0, 0), H_ - 1);
    const int cy1 = min(max(y1, 0), H_ - 1);
    const int cx0 = min(max(x0, 0), W_ - 1);
    const int cx1 = min(max(x1, 0), W_ - 1);
    const float4* r00 = (const float4*)(xb + ((size_t)cy0 * W_ + cx0) * C_ + c0);
    const float4* r01 = (const float4*)(xb + ((size_t)cy0 * W_ + cx1) * C_ + c0);
    const float4* r10 = (const float4*)(xb + ((size_t)cy1 * W_ + cx0) * C_ + c0);
    const float4* r11 = (const float4*)(xb + ((size_t)cy1 * W_ + cx1) * C_ + c0);
#pragma unroll
    for (int q = 0; q < 8; ++q) {
      const float4 a0 = r00[q], a1 = r01[q], a2 = r10[q], a3 = r11[q];
      const int cp = (c0 + q * 4) >> 1;  // even pair index
      v2f t01, t23;
      t01.x = w00 * a0.x + w01 * a1.x + w10 * a2.x + w11 * a3.x;
      t01.y = w00 * a0.y + w01 * a1.y + w10 * a2.y + w11 * a3.y;
      t23.x = w00 * a0.z + w01 * a1.z + w10 * a2.z + w11 * a3.z;
      t23.y = w00 * a0.w + w01 * a1.w + w10 * a2.w + w11 * a3.w;
      *(v2f*)&val[cp * PP + pid * 2]       = t01;
      *(v2f*)&val[(cp + 1) * PP + pid * 2] = t23;
    }
    __syncthreads();

    // ---- WMMA phase: acc += W_tap[16oc x 128c] * val[128c x 64px] ----
    // A frag (16x4 f32): lane L, vgpr j -> (M = L%16, K = j + 2*(L>=16))
    // B frag (4x16 f32): lane L, vgpr j -> (N = L%16, K = j + 2*(L>=16))
    // K-pair {ka, ka+1} = pair row cp = kc/2 + hi in both repacked layouts.
#if defined(__HIP_DEVICE_COMPILE__)
    const float* wk = w_t + k * C_ * O_;
    for (int kc = 0; kc < C_; kc += 4) {
      const int cp = (kc >> 1) + hi;
      const v2f a = *(const v2f*)(wk + (size_t)cp * (2 * O_) + (ocBase + lo) * 2);
#pragma unroll
      for (int s = 0; s < 4; ++s) {
        const v2f bb = *(const v2f*)&val[cp * PP + (s * 16 + lo) * 2];
        acc[s] = __builtin_amdgcn_wmma_f32_16x16x4_f32(
            false, a, false, bb, (short)0, acc[s], false, false);
      }
    }
#endif
    __syncthreads();
  }

  // ---- epilogue: C/D layout lane L, vgpr v -> (N = L%16, M = v + 8*(L>=16))
#pragma unroll
  for (int s = 0; s < 4; ++s) {
    const int p = pixBase + s * 16 + lo;
#pragma unroll
    for (int v = 0; v < 8; ++v) {
      const int oc = ocBase + v + 8 * hi;
      out[((size_t)b * O_ + oc) * HW_ + p] = acc[s][v] + db[oc];
    }
  }
}

// ---------------------------------------------------------------------------
extern "C" void kernel_launch(void* const* d_in, const int* in_sizes, int n_in,
                              void* d_out, int out_size, void* d_ws,
                              size_t ws_size, hipStream_t stream) {
  const float* x        = (const float*)d_in[0];
  const float* x2       = (const float*)d_in[1];
  const float* offset_w = (const float*)d_in[2];
  const float* offset_b = (const float*)d_in[3];
  const float* mask_w   = (const float*)d_in[4];
  const float* mask_b   = (const float*)d_in[5];
  const float* deform_w = (const float*)d_in[6];
  const float* deform_b = (const float*)d_in[7];
  float* out = (float*)d_out;

  // workspace layout (floats); total ~= 42.5 MB
  float* ws      = (float*)d_ws;
  float* x_t     = ws;                              // B*HW*C
  float* x2_t    = x_t + (size_t)B_ * HW_ * C_;     // B*HW*C
  float* w_t     = x2_t + (size_t)B_ * HW_ * C_;    // K*(C/2)*O*2
  float* w1_t    = w_t + (size_t)K_ * C_ * O_;      // K*C*27
  float* offmask = w1_t + (size_t)K_ * C_ * NCH;    // B*27*HW

  const dim3 tb(32, 8, 1);
  transpose_chw_hwc<<<dim3(HW_ / 32, C_ / 32, B_), tb, 0, stream>>>(x, x_t);
  transpose_chw_hwc<<<dim3(HW_ / 32, C_ / 32, B_), tb, 0, stream>>>(x2, x2_t);

  const int nw = K_ * C_ * O_;
  reorg_weights<<<(nw + 255) / 256, 256, 0, stream>>>(deform_w, offset_w,
                                                      mask_w, w_t, w1_t);

  offset_mask_conv<<<B_ * (HW_ / 256), 256, 0, stream>>>(x2_t, w1_t, offset_b,
                                                         mask_b, offmask);

  deform_wmma<<<B_ * (HW_ / 64), 256, 0, stream>>>(x_t, offmask, w_t,
                                                   deform_b, out);
}